// EdgeEncoding_13855564497055
// MI455X (gfx1250) — compile-verified
//
#include <hip/hip_runtime.h>

typedef __attribute__((ext_vector_type(2))) float v2f;
typedef __attribute__((ext_vector_type(8))) float v8f;

#define DDIM 32            // edge feature dim (reference D=32)
#define KCHUNKS (DDIM / 4) // 8 x K=4 WMMA steps cover D=32
#define DSTRIDE 8          // padded row stride of the dots table (requires L<=8)
#define BLK 256            // 8 waves per block (wave32)

// ---------------------------------------------------------------------------
// Kernel 1: dots[e][l] = dot(edge_attr[e,:], edge_vector[l,:])
// Computed as (E x 32) * (32 x 16) GEMM tiles via V_WMMA_F32_16X16X4_F32,
// with B columns l >= L zero-padded. One wave handles 16 edge rows.
// ---------------------------------------------------------------------------
__global__ void __launch_bounds__(BLK)
edge_dots_wmma(const float* __restrict__ edge_attr,
               const float* __restrict__ edge_vector,
               float* __restrict__ dots, int E, int L) {
  const int wave = (int)((blockIdx.x * BLK + threadIdx.x) >> 5);
  const int lane = (int)(threadIdx.x & 31);
  const int e0 = wave * 16;
  if (e0 >= E) return;               // wave-uniform: EXEC stays all-1s for WMMA

  const int half = lane >> 4;        // 0: lanes 0-15, 1: lanes 16-31
  const int m = lane & 15;           // A row within tile; also B/D column index

  // B-matrix (4x16, K x N): VGPR0 holds K=kb (lanes 0-15) / K=kb+2 (16-31),
  // VGPR1 holds K=kb+1 / kb+3.  B[k][n] = edge_vector[n][k], n = m, zero-pad n>=L.
  v2f b[KCHUNKS];
#pragma unroll
  for (int c = 0; c < KCHUNKS; ++c) {
    const int kb = c * 4 + half * 2;
    float b0 = 0.0f, b1 = 0.0f;
    if (m < L) {
      b0 = edge_vector[m * DDIM + kb];
      b1 = edge_vector[m * DDIM + kb + 1];
    }
    b[c].x = b0;
    b[c].y = b1;
  }

  // A-matrix (16x4): lane m holds A[m][kb], A[m][kb+1]
  const float* arow = edge_attr + (size_t)(e0 + m) * DDIM + half * 2;
  v8f acc = {};
#pragma unroll
  for (int c = 0; c < KCHUNKS; ++c) {
    v2f a;
    a.x = arow[c * 4 + 0];
    a.y = arow[c * 4 + 1];
    // 8 args: (neg_a, A, neg_b, B, c_mod, C, reuse_a, reuse_b)
    acc = __builtin_amdgcn_wmma_f32_16x16x4_f32(false, a, false, b[c],
                                                (short)0, acc, false, false);
  }

  // D layout: VGPR r -> row = half*8 + r, col n = m. Store only real columns.
  if (m < L) {
#pragma unroll
    for (int r = 0; r < 8; ++r) {
      const int row = half * 8 + r;
      dots[(size_t)(e0 + row) * DSTRIDE + m] = acc[r];
    }
  }
}

// ---------------------------------------------------------------------------
// Kernel 2: per pair p, masked mean of dots[path_edges[p][l]][l] over l<len.
// path_edges rows staged through LDS so global loads are fully coalesced;
// the dots table (4 MB) is L2-resident on MI455X (192 MB L2).
// ---------------------------------------------------------------------------
__global__ void __launch_bounds__(BLK)
path_mean(const int* __restrict__ path_edges,
          const int* __restrict__ path_len,
          const float* __restrict__ dots,
          float* __restrict__ out, int P, int L) {
  __shared__ int s_edges[BLK * DSTRIDE];  // 8 KB of the 320 KB/WGP LDS

  const int p0 = (int)blockIdx.x * BLK;
  const int total = BLK * L;
  const long long base = (long long)p0 * L;
  const long long limit = (long long)P * L;
  for (int i = (int)threadIdx.x; i < total; i += BLK) {
    const long long g = base + i;
    s_edges[i] = (g < limit) ? path_edges[g] : 0;
  }
  __syncthreads();

  const int p = p0 + (int)threadIdx.x;
  if (p >= P) return;

  const int len = path_len[p];   // reference guarantees 0 <= len <= L <= DSTRIDE
  float s = 0.0f;
#pragma unroll
  for (int l = 0; l < DSTRIDE; ++l) {
    if (l < len) {
      const int e = s_edges[(int)threadIdx.x * L + l];
      s += dots[(size_t)e * DSTRIDE + l];
    }
  }
  out[p] = (len > 0) ? (s / (float)len) : 0.0f;
}

// ---------------------------------------------------------------------------
// Host launcher. Inputs (setup_inputs order):
//   d_in[0] x           (N*D f32)   -- unused by the math
//   d_in[1] edge_attr   (E*D f32)
//   d_in[2] edge_vector (L*D f32)
//   d_in[3] path_edges  (N*N*L i32)
//   d_in[4] path_len    (N*N   i32)
// d_out: N*N f32.  d_ws: dots table, E*DSTRIDE*4 = 4 MB.
// ---------------------------------------------------------------------------
extern "C" void kernel_launch(void* const* d_in, const int* in_sizes, int n_in,
                              void* d_out, int out_size, void* d_ws, size_t ws_size,
                              hipStream_t stream) {
  const float* edge_attr   = (const float*)d_in[1];
  const float* edge_vector = (const float*)d_in[2];
  const int*   path_edges  = (const int*)d_in[3];
  const int*   path_len    = (const int*)d_in[4];
  float*       out         = (float*)d_out;

  const int P = in_sizes[4];                 // N*N pairs
  const int L = in_sizes[3] / P;             // max path length (5)
  const int D = in_sizes[2] / L;             // edge dim (32); WMMA path assumes 32
  const int E = in_sizes[1] / D;             // number of edges
  (void)D; (void)n_in; (void)out_size; (void)ws_size;

  float* dots = (float*)d_ws;                // E * DSTRIDE floats

  // Kernel 1: one wave per 16 edges, 8 waves per 256-thread block.
  const int waves1  = (E + 15) / 16;
  const int blocks1 = (waves1 + (BLK / 32) - 1) / (BLK / 32);
  edge_dots_wmma<<<blocks1, BLK, 0, stream>>>(edge_attr, edge_vector, dots, E, L);

  // Kernel 2: one thread per pair.
  const int blocks2 = (P + BLK - 1) / BLK;
  path_mean<<<blocks2, BLK, 0, stream>>>(path_edges, path_len, dots, out, P, L);
}